// NNDModule_7164005450406
// MI455X (gfx1250) — compile-verified
//
#include <hip/hip_runtime.h>
#include <math.h>

typedef __attribute__((ext_vector_type(2))) float v2f;
typedef __attribute__((ext_vector_type(4))) float v4f;
typedef __attribute__((ext_vector_type(8))) float v8f;

#define NPTS       16384
#define BATCH      8
#define CHUNK_PTS  2048
#define BLOCK_THREADS 256
#define QPW        32            // queries per wave: two 16-col B tiles
#define QPB        (8 * QPW)     // 256 queries per block

// One direction of Chamfer NN. For each query x_j, min_i |x_j - y_i|^2.
// The K=3 pad slot of the 16x16x4 f32 WMMA carries the norms:
//   A = (-2y.x, -2y.y, -2y.z, |y|^2)   (db tile, 16x4)
//   B = ( x.x,   x.y,   x.z,  1    )   (query tile, 4x16)
//   D = A x B + 0 = |y_i|^2 - 2 x_j . y_i      (C is inline 0)
// dist_j = min_i D_ij + |x_j|^2 (added once at the end).
// Each wave keeps TWO query tiles resident; each iteration processes TWO
// A-tiles (one ds_load_2addr_b64) -> 4 WMMAs, and the reduction is written
// as fminf(acc, fminf(dA,dB)) so each accumulator update is one v_min3.
// Unroll kept at 1 so only 4 WMMA results (32 VGPRs) are in flight,
// avoiding the register-pressure canonicalize copies seen with deeper unroll.
__global__ __launch_bounds__(BLOCK_THREADS)
void chamfer_nn_dir(const float* __restrict__ qry,
                    const float* __restrict__ db,
                    float* __restrict__ out)
{
    __shared__ __align__(16) float ldsY[CHUNK_PTS * 4]; // (-2x,-2y,-2z,|y|^2)

    const int lane = threadIdx.x & 31;
    const int wave = threadIdx.x >> 5;
    const int blocksPerBatch = NPTS / QPB;              // 64
    const int b    = blockIdx.x / blocksPerBatch;
    const int qblk = blockIdx.x % blocksPerBatch;
    const int qbase = qblk * QPB + wave * QPW;

    // ---- Two B matrices (query tiles). Lanes 0-15 carry K=0,1; lanes 16-31
    // carry K=2,3 with the K=3 row = 1.0. Column j = lane % 16.
    const int j = lane & 15;
    const float* qp0 = qry + ((size_t)b * NPTS + qbase + j) * 3;
    const float* qp1 = qp0 + 16 * 3;
    v2f bmat0, bmat1;
    float part0, part1;
    if (lane < 16) {
        bmat0.x = qp0[0]; bmat0.y = qp0[1];
        bmat1.x = qp1[0]; bmat1.y = qp1[1];
        part0 = bmat0.x * bmat0.x + bmat0.y * bmat0.y;
        part1 = bmat1.x * bmat1.x + bmat1.y * bmat1.y;
    } else {
        bmat0.x = qp0[2]; bmat0.y = 1.0f;
        bmat1.x = qp1[2]; bmat1.y = 1.0f;
        part0 = bmat0.x * bmat0.x;
        part1 = bmat1.x * bmat1.x;
    }
    float x2_0 = part0 + __shfl_xor(part0, 16, 32);
    float x2_1 = part1 + __shfl_xor(part1, 16, 32);

    float rmin0[8], rmin1[8];
#pragma unroll
    for (int v = 0; v < 8; ++v) { rmin0[v] = INFINITY; rmin1[v] = INFINITY; }

    const float* dbB = db + (size_t)b * NPTS * 3;
    const int aOff = j * 4 + ((lane >= 16) ? 2 : 0); // ds_load_b64 -> A vgpr pair
    const v8f czero = {};

    for (int chunk = 0; chunk < NPTS; chunk += CHUNK_PTS) {
        __syncthreads();
        // Cooperative stage: transform db points into WMMA A-operand layout.
#pragma unroll
        for (int k = 0; k < CHUNK_PTS / BLOCK_THREADS; ++k) {
            int p = threadIdx.x + k * BLOCK_THREADS;
            const float* src = dbB + (size_t)(chunk + p) * 3;
            float px = src[0], py = src[1], pz = src[2];
            v4f pk;
            pk.x = -2.0f * px;
            pk.y = -2.0f * py;
            pk.z = -2.0f * pz;
            pk.w = px * px + py * py + pz * pz;   // |y|^2 rides in the K=3 slot
            *(v4f*)&ldsY[p * 4] = pk;
        }
        __syncthreads();

        // 64 iterations, each: 2 A-tiles x 2 query tiles = 1024 pairs.
#pragma unroll 1
        for (int t = 0; t < CHUNK_PTS / 16; t += 2) {
            v2f a0 = *(const v2f*)&ldsY[t * 64 + aOff];
            v2f a1 = *(const v2f*)&ldsY[t * 64 + 64 + aOff];
            v8f d00 = __builtin_amdgcn_wmma_f32_16x16x4_f32(
                false, a0, false, bmat0, (short)0, czero, false, false);
            v8f d10 = __builtin_amdgcn_wmma_f32_16x16x4_f32(
                false, a1, false, bmat0, (short)0, czero, false, false);
            v8f d01 = __builtin_amdgcn_wmma_f32_16x16x4_f32(
                false, a0, false, bmat1, (short)0, czero, false, false);
            v8f d11 = __builtin_amdgcn_wmma_f32_16x16x4_f32(
                false, a1, false, bmat1, (short)0, czero, false, false);
#pragma unroll
            for (int v = 0; v < 8; ++v) {
                rmin0[v] = fminf(rmin0[v], fminf(d00[v], d10[v])); // v_min3
                rmin1[v] = fminf(rmin1[v], fminf(d01[v], d11[v])); // v_min3
            }
        }
    }

    // Reduce 8 rows in-lane, then across the two lane halves (rows 0-7 vs 8-15).
    float m0 = rmin0[0], m1 = rmin1[0];
#pragma unroll
    for (int v = 1; v < 8; ++v) { m0 = fminf(m0, rmin0[v]); m1 = fminf(m1, rmin1[v]); }
    m0 = fminf(m0, __shfl_xor(m0, 16, 32));
    m1 = fminf(m1, __shfl_xor(m1, 16, 32));
    if (lane < 16) {
        out[(size_t)b * NPTS + qbase + j]      = m0 + x2_0;
        out[(size_t)b * NPTS + qbase + 16 + j] = m1 + x2_1;
    }
}

extern "C" void kernel_launch(void* const* d_in, const int* in_sizes, int n_in,
                              void* d_out, int out_size, void* d_ws, size_t ws_size,
                              hipStream_t stream) {
    const float* x = (const float*)d_in[0];
    const float* y = (const float*)d_in[1];
    float* out = (float*)d_out;

    dim3 grid(BATCH * (NPTS / QPB));   // 512 blocks per direction
    dim3 block(BLOCK_THREADS);

    // dist1: queries = input1, database = input2
    chamfer_nn_dir<<<grid, block, 0, stream>>>(x, y, out);
    // dist2: queries = input2, database = input1
    chamfer_nn_dir<<<grid, block, 0, stream>>>(y, x, out + (size_t)BATCH * NPTS);
}